// GCNEmbedder_32684701122868
// MI455X (gfx1250) — compile-verified
//
#include <hip/hip_runtime.h>

#define N_NODES 100000
#define N_EDGES 1600000
#define HID     128
#define N_GRAPH 1024

typedef __attribute__((ext_vector_type(2))) float v2f;
typedef __attribute__((ext_vector_type(8))) float v8f;

__device__ __forceinline__ void atomAddF(float* p, float v) {
    __hip_atomic_fetch_add(p, v, __ATOMIC_RELAXED, __HIP_MEMORY_SCOPE_AGENT);
}

// ---------------- degree / normalization ----------------
__global__ __launch_bounds__(256) void k_deg_init(float* deg) {
    int i = blockIdx.x * 256 + threadIdx.x;
    if (i < N_NODES) deg[i] = 1.0f;                 // self loop contributes 1
}

__global__ __launch_bounds__(256) void k_deg_acc(const int* __restrict__ ei, float* deg) {
    int e = blockIdx.x * 256 + threadIdx.x;
    if (e < N_EDGES) atomAddF(&deg[ei[N_EDGES + e]], 1.0f);   // dst row
}

__global__ __launch_bounds__(256) void k_deg_inv(float* deg) {
    int i = blockIdx.x * 256 + threadIdx.x;
    if (i < N_NODES) deg[i] = rsqrtf(deg[i]);       // deg >= 1 always
}

// ---------------- embedding gather ----------------
__global__ __launch_bounds__(256) void k_embed(const int* __restrict__ x,
                                               const float* __restrict__ emb,
                                               float* __restrict__ h) {
    int idx  = blockIdx.x * 256 + threadIdx.x;      // N*32 threads
    int n    = idx >> 5;
    int c4   = (idx & 31) * 4;
    float4 v = *(const float4*)(emb + (size_t)x[n] * HID + c4);
    *(float4*)(h + (size_t)n * HID + c4) = v;
}

// ---------------- WMMA fp32 GEMM: B = (relu?)A @ W ----------------
// block = 256 threads = 8 waves; block b covers rows [16b,16b+16), wave w covers cols [16w,16w+16)
template <int RELU>
__global__ __launch_bounds__(256) void k_gemm_wmma(const float* __restrict__ A,
                                                   const float* __restrict__ W,
                                                   float* __restrict__ B) {
    const int lane = threadIdx.x & 31;
    const int wave = threadIdx.x >> 5;
    const int row0 = blockIdx.x * 16;
    const int col0 = wave * 16;
    const int m    = lane & 15;        // row within A tile / col within B tile
    const int kh   = lane >> 4;        // 0 or 1: K-half selector

    v8f acc = {0.f, 0.f, 0.f, 0.f, 0.f, 0.f, 0.f, 0.f};
    const float* arow = A + (size_t)(row0 + m) * HID;

    #pragma unroll 4
    for (int k = 0; k < HID; k += 4) {
        int ka = k + 2 * kh;
        float a0 = arow[ka];
        float a1 = arow[ka + 1];
        if (RELU) { a0 = fmaxf(a0, 0.f); a1 = fmaxf(a1, 0.f); }
        v2f a; a.x = a0; a.y = a1;
        v2f b;
        b.x = W[(size_t)ka * HID + col0 + m];
        b.y = W[(size_t)(ka + 1) * HID + col0 + m];
        acc = __builtin_amdgcn_wmma_f32_16x16x4_f32(false, a, false, b,
                                                    (short)0, acc, false, false);
    }
    // C/D layout: VGPR g, lanes 0-15 -> row row0+g; lanes 16-31 -> row row0+8+g
    float* out = B + (size_t)(row0 + kh * 8) * HID + col0 + m;
    #pragma unroll
    for (int g = 0; g < 8; ++g) out[(size_t)g * HID] = acc[g];
}

// ---------------- self-loop + bias init: hout = bias + dinv^2 * hw ----------------
__global__ __launch_bounds__(256) void k_self_init(const float* __restrict__ hw,
                                                   float* __restrict__ hout,
                                                   const float* __restrict__ dinv,
                                                   const float* __restrict__ bias) {
    int idx = blockIdx.x * 256 + threadIdx.x;       // N*32 threads
    int n   = idx >> 5;
    int c4  = (idx & 31) * 4;
    float di = dinv[n];
    float nn = di * di;
    float4 v = *(const float4*)(hw + (size_t)n * HID + c4);
    float4 b = *(const float4*)(bias + c4);
    float4 r;
    r.x = b.x + nn * v.x;  r.y = b.y + nn * v.y;
    r.z = b.z + nn * v.z;  r.w = b.w + nn * v.w;
    *(float4*)(hout + (size_t)n * HID + c4) = r;
}

// ---------------- edge scatter: hout[dst] += norm * hw[src]; one wave per edge ----------------
__global__ __launch_bounds__(256) void k_edge_agg(const float* __restrict__ hw,
                                                  float* __restrict__ hout,
                                                  const int* __restrict__ ei,
                                                  const float* __restrict__ dinv) {
    int gtid = blockIdx.x * 256 + threadIdx.x;      // E*32 threads
    int e    = gtid >> 5;
    int lane = gtid & 31;
    int s = ei[e];
    int d = ei[N_EDGES + e];
    float norm = dinv[s] * dinv[d];
    float4 v = *(const float4*)(hw + (size_t)s * HID + lane * 4);
    float* dst = hout + (size_t)d * HID + lane * 4;
    atomAddF(dst + 0, v.x * norm);
    atomAddF(dst + 1, v.y * norm);
    atomAddF(dst + 2, v.z * norm);
    atomAddF(dst + 3, v.w * norm);
}

// ---------------- pooling ----------------
__global__ __launch_bounds__(256) void k_pool_zero(float* out, float* cnt) {
    int i = blockIdx.x * 256 + threadIdx.x;
    if (i < N_GRAPH * HID) out[i] = 0.f;
    if (i < N_GRAPH) cnt[i] = 0.f;
}

__global__ __launch_bounds__(256) void k_pool_acc(const float* __restrict__ h,
                                                  const int* __restrict__ batch,
                                                  float* __restrict__ out,
                                                  float* __restrict__ cnt) {
    int idx  = blockIdx.x * 256 + threadIdx.x;      // N*32 threads
    int n    = idx >> 5;
    int lane = idx & 31;
    int g    = batch[n];
    float4 v = *(const float4*)(h + (size_t)n * HID + lane * 4);
    float* o = out + (size_t)g * HID + lane * 4;
    atomAddF(o + 0, v.x);
    atomAddF(o + 1, v.y);
    atomAddF(o + 2, v.z);
    atomAddF(o + 3, v.w);
    if (lane == 0) atomAddF(&cnt[g], 1.0f);
}

__global__ __launch_bounds__(256) void k_pool_div(float* out, const float* __restrict__ cnt) {
    int i = blockIdx.x * 256 + threadIdx.x;
    if (i < N_GRAPH * HID) out[i] /= fmaxf(cnt[i >> 7], 1.0f);
}

extern "C" void kernel_launch(void* const* d_in, const int* in_sizes, int n_in,
                              void* d_out, int out_size, void* d_ws, size_t ws_size,
                              hipStream_t stream) {
    const int*   x     = (const int*)d_in[0];
    const int*   ei    = (const int*)d_in[1];     // [2,E]
    const int*   batch = (const int*)d_in[2];
    const float* emb   = (const float*)d_in[3];   // [V,H]
    const float* Ws    = (const float*)d_in[4];   // [L,H,H]
    const float* bs    = (const float*)d_in[5];   // [L,H]
    float* out = (float*)d_out;

    float* ws   = (float*)d_ws;
    float* dinv = ws;                                   // N floats
    float* bufA = ws + N_NODES;                         // N*H floats
    float* bufB = bufA + (size_t)N_NODES * HID;         // N*H floats
    float* cnt  = bufB + (size_t)N_NODES * HID;         // G floats

    const int NT = 256;
    // normalization
    k_deg_init<<<(N_NODES + NT - 1) / NT, NT, 0, stream>>>(dinv);
    k_deg_acc <<<(N_EDGES + NT - 1) / NT, NT, 0, stream>>>(ei, dinv);
    k_deg_inv <<<(N_NODES + NT - 1) / NT, NT, 0, stream>>>(dinv);
    // embedding lookup -> bufA
    k_embed<<<(N_NODES * 32) / NT, NT, 0, stream>>>(x, emb, bufA);
    // 3 GCN layers
    for (int l = 0; l < 3; ++l) {
        if (l == 0)
            k_gemm_wmma<0><<<N_NODES / 16, NT, 0, stream>>>(bufA, Ws, bufB);
        else
            k_gemm_wmma<1><<<N_NODES / 16, NT, 0, stream>>>(bufA, Ws + (size_t)l * HID * HID, bufB);
        k_self_init<<<(N_NODES * 32) / NT, NT, 0, stream>>>(bufB, bufA, dinv,
                                                            bs + (size_t)l * HID);
        k_edge_agg<<<(N_EDGES * 32) / NT, NT, 0, stream>>>(bufB, bufA, ei, dinv);
    }
    // mean pooling
    k_pool_zero<<<(N_GRAPH * HID + NT - 1) / NT, NT, 0, stream>>>(out, cnt);
    k_pool_acc <<<(N_NODES * 32) / NT, NT, 0, stream>>>(bufA, batch, out, cnt);
    k_pool_div <<<(N_GRAPH * HID + NT - 1) / NT, NT, 0, stream>>>(out, cnt);
}